// GraphAttentionLayer_61804579389526
// MI455X (gfx1250) — compile-verified
//
#include <hip/hip_runtime.h>
#include <hip/hip_bf16.h>

#define N_NODES 10000
#define DIM 512
#define NHEAD 8
#define HDIM 64
#define NEDGE 160000
#define ETOT (NEDGE + N_NODES)
#define NEG_SLOPE 0.2f
#define BN_EPS 1e-5f

typedef float v2f __attribute__((ext_vector_type(2)));
typedef float v8f __attribute__((ext_vector_type(8)));

// ---------------------------------------------------------------------------
// GEMM: out[10000 x 512] = X[10000 x 512] @ W[512 x 512] (+ optional bias)
// 256 threads = 8 wave32; block tile 16 rows x 128 cols; wave tile 16x16.
// K staged through LDS in 64-wide chunks. Uses V_WMMA_F32_16X16X4_F32.
// ISA operand layout (CDNA5 ISA):
//   A 16x4 f32: lane<16 -> {A[r][k0],A[r][k0+1]}, lane>=16 -> {k0+2,k0+3}
//   B 4x16 f32: lane<16 -> {B[k0][c],B[k0+1][c]}, lane>=16 -> {k0+2,k0+3}
//   C/D: VGPR r -> lanes 0-15 M=r, lanes 16-31 M=r+8
// ---------------------------------------------------------------------------
__global__ __launch_bounds__(256) void gemm_wmma_f32(
    const float* __restrict__ X, const float* __restrict__ W,
    const float* __restrict__ bias, float* __restrict__ out)
{
    __shared__ float As[16 * 68];    // 16 rows x 64 k, pitch 68 (bank spread)
    __shared__ float Bs[64 * 136];   // 64 k x 128 cols, pitch 136 (half-wave banks disjoint)

    const int rowBase = blockIdx.x * 16;
    const int colBase = blockIdx.y * 128;
    const int t    = threadIdx.x;
    const int wave = t >> 5;
    const int lane = t & 31;
    const int half = lane >> 4;
    const int lr   = lane & 15;

    v8f acc = {};

    for (int kBase = 0; kBase < DIM; kBase += 64) {
        // stage A chunk (16 x 64): one float4 per thread
        {
            const int r  = t >> 4;
            const int c4 = (t & 15) << 2;
            const float4 av = *(const float4*)(X + (size_t)(rowBase + r) * DIM + kBase + c4);
            *(float4*)(As + r * 68 + c4) = av;
        }
        // stage B chunk (64 x 128): 8 float4 per thread
        #pragma unroll
        for (int i = 0; i < 8; ++i) {
            const int f4 = i * 256 + t;
            const int r  = f4 >> 5;
            const int c4 = (f4 & 31) << 2;
            const float4 bv = *(const float4*)(W + (size_t)(kBase + r) * DIM + colBase + c4);
            *(float4*)(Bs + r * 136 + c4) = bv;
        }
        __syncthreads();

        #pragma unroll
        for (int kk = 0; kk < 64; kk += 4) {
            const v2f a = *(const v2f*)(As + lr * 68 + kk + 2 * half);
            v2f b;
            const int bo = (kk + 2 * half) * 136 + wave * 16 + lr;
            b.x = Bs[bo];
            b.y = Bs[bo + 136];
            acc = __builtin_amdgcn_wmma_f32_16x16x4_f32(
                false, a, false, b, (short)0, acc, false, false);
        }
        __syncthreads();
    }

    const int col  = colBase + wave * 16 + lr;
    const float ba = bias ? bias[col] : 0.0f;
    #pragma unroll
    for (int r = 0; r < 8; ++r) {
        const int row = rowBase + r + 8 * half;
        out[(size_t)row * DIM + col] = acc[r] + ba;
    }
}

// a_src[n,h] = <h[n,h,:], att_src[h,:]> ; same for a_dst
__global__ void att_scores(const float* __restrict__ hfeat,
                           const float* __restrict__ att_src,
                           const float* __restrict__ att_dst,
                           float* __restrict__ a_src, float* __restrict__ a_dst)
{
    const int idx = blockIdx.x * blockDim.x + threadIdx.x;
    if (idx >= N_NODES * NHEAD) return;
    const int n  = idx >> 3;
    const int hh = idx & 7;
    const float* hp = hfeat + (size_t)n * DIM + hh * HDIM;
    const float* as = att_src + hh * HDIM;
    const float* ad = att_dst + hh * HDIM;
    float ss = 0.f, sd = 0.f;
    #pragma unroll 8
    for (int c = 0; c < HDIM; ++c) {
        const float v = hp[c];
        ss += v * as[c];
        sd += v * ad[c];
    }
    a_src[idx] = ss;
    a_dst[idx] = sd;
}

// per-edge: ex = exp(leaky_relu(a_src[src] + a_dst[dst])); denom[dst] += ex
// (softmax shift-invariance => no segment-max pass needed)
__global__ void edge_exp(const int* __restrict__ ei,
                         const float* __restrict__ a_src,
                         const float* __restrict__ a_dst,
                         float* __restrict__ exbuf, float* __restrict__ denom)
{
    const int i = blockIdx.x * blockDim.x + threadIdx.x;
    if (i >= ETOT) return;
    int s, d;
    if (i < NEDGE) { s = ei[i]; d = ei[NEDGE + i]; }
    else           { s = d = i - NEDGE; }
    #pragma unroll
    for (int hh = 0; hh < NHEAD; ++hh) {
        float e = a_src[s * NHEAD + hh] + a_dst[d * NHEAD + hh];
        e = (e >= 0.f) ? e : NEG_SLOPE * e;
        const float ex = expf(e);
        exbuf[(size_t)i * NHEAD + hh] = ex;
        atomicAdd(&denom[d * NHEAD + hh], ex);
    }
}

// one block per edge: out[dst,:] += alpha * h[src,:]
__global__ __launch_bounds__(128) void edge_scatter(
    const int* __restrict__ ei, const float* __restrict__ hfeat,
    const float* __restrict__ exbuf, const float* __restrict__ denom,
    float* __restrict__ outacc)
{
    const int i = blockIdx.x;
    int s, d;
    if (i < NEDGE) { s = ei[i]; d = ei[NEDGE + i]; }
    else           { s = d = i - NEDGE; }
    const int j  = threadIdx.x << 2;        // 0..508
    const int hh = j >> 6;                  // head for this 64-chunk
    const float alpha = exbuf[(size_t)i * NHEAD + hh] / denom[d * NHEAD + hh];
    const float4 hv = *(const float4*)(hfeat + (size_t)s * DIM + j);
    float* op = outacc + (size_t)d * DIM + j;
    atomicAdd(op + 0, alpha * hv.x);
    atomicAdd(op + 1, alpha * hv.y);
    atomicAdd(op + 2, alpha * hv.z);
    atomicAdd(op + 3, alpha * hv.w);
}

// column sums / sums of squares (coalesced: thread = column)
__global__ __launch_bounds__(512) void col_stats(const float* __restrict__ outacc,
                                                 float* __restrict__ colsum,
                                                 float* __restrict__ colsq)
{
    const int col = threadIdx.x;
    const int r0  = blockIdx.x * 250;
    float s = 0.f, q = 0.f;
    for (int r = r0; r < r0 + 250; ++r) {
        const float v = outacc[(size_t)r * DIM + col];
        s += v;
        q += v * v;
    }
    atomicAdd(&colsum[col], s);
    atomicAdd(&colsq[col], q);
}

// fold BN into per-column scale/shift
__global__ void bn_prep(const float* __restrict__ colsum, const float* __restrict__ colsq,
                        const float* __restrict__ gamma, const float* __restrict__ beta,
                        float* __restrict__ scale, float* __restrict__ shift)
{
    const int j = threadIdx.x;
    const float mean = colsum[j] * (1.0f / N_NODES);
    const float var  = colsq[j] * (1.0f / N_NODES) - mean * mean;
    const float sc   = gamma[j] / sqrtf(var + BN_EPS);
    scale[j] = sc;
    shift[j] = beta[j] - mean * sc;
}

// out = gelu_erf(BN(outacc)) + identity
__global__ void bn_gelu_add(const float* __restrict__ outacc,
                            const float* __restrict__ scale,
                            const float* __restrict__ shift,
                            const float* __restrict__ ident,
                            float* __restrict__ out)
{
    const int idx = blockIdx.x * blockDim.x + threadIdx.x;
    if (idx >= N_NODES * DIM) return;
    const int j = idx & (DIM - 1);
    const float xn = outacc[idx] * scale[j] + shift[j];
    const float g  = 0.5f * xn * (1.0f + erff(xn * 0.70710678118654752f));
    out[idx] = g + ident[idx];
}

extern "C" void kernel_launch(void* const* d_in, const int* in_sizes, int n_in,
                              void* d_out, int out_size, void* d_ws, size_t ws_size,
                              hipStream_t stream)
{
    const float* x        = (const float*)d_in[0];
    const int*   ei       = (const int*)d_in[1];
    const float* W_gat    = (const float*)d_in[2];
    const float* att_src  = (const float*)d_in[3];
    const float* att_dst  = (const float*)d_in[4];
    // d_in[5] = gat_bias: per-column shift, cancels exactly in BatchNorm -> skipped
    const float* bn_gamma = (const float*)d_in[6];
    const float* bn_beta  = (const float*)d_in[7];
    const float* W_skip   = (const float*)d_in[8];
    const float* b_skip   = (const float*)d_in[9];
    float* out = (float*)d_out;

    float* ws = (float*)d_ws;
    float* h_feat = ws;  ws += (size_t)N_NODES * DIM;      // 5.12M
    float* ident  = ws;  ws += (size_t)N_NODES * DIM;      // 5.12M
    float* a_src  = ws;  ws += N_NODES * NHEAD;            // 80k
    float* a_dst  = ws;  ws += N_NODES * NHEAD;            // 80k
    float* exbuf  = ws;  ws += (size_t)ETOT * NHEAD;       // 1.36M
    float* denom  = ws;  ws += N_NODES * NHEAD;            // 80k
    float* outacc = ws;  ws += (size_t)N_NODES * DIM;      // 5.12M
    float* colsum = ws;  ws += DIM;
    float* colsq  = ws;  ws += DIM;
    float* scale  = ws;  ws += DIM;
    float* shift  = ws;  ws += DIM;

    // per-call zeroing of accumulators (graph-capture safe)
    hipMemsetAsync(denom,  0, (size_t)N_NODES * NHEAD * sizeof(float), stream);
    hipMemsetAsync(outacc, 0, (size_t)N_NODES * DIM * sizeof(float), stream);
    hipMemsetAsync(colsum, 0, 2 * DIM * sizeof(float), stream);

    const dim3 ggrid(N_NODES / 16, DIM / 128);
    gemm_wmma_f32<<<ggrid, 256, 0, stream>>>(x, W_gat, nullptr, h_feat);
    gemm_wmma_f32<<<ggrid, 256, 0, stream>>>(x, W_skip, b_skip, ident);

    att_scores<<<(N_NODES * NHEAD + 255) / 256, 256, 0, stream>>>(
        h_feat, att_src, att_dst, a_src, a_dst);
    edge_exp<<<(ETOT + 255) / 256, 256, 0, stream>>>(ei, a_src, a_dst, exbuf, denom);
    edge_scatter<<<ETOT, 128, 0, stream>>>(ei, h_feat, exbuf, denom, outacc);
    col_stats<<<N_NODES / 250, 512, 0, stream>>>(outacc, colsum, colsq);
    bn_prep<<<1, DIM, 0, stream>>>(colsum, colsq, bn_gamma, bn_beta, scale, shift);
    bn_gelu_add<<<(N_NODES * DIM + 255) / 256, 256, 0, stream>>>(
        outacc, scale, shift, ident, out);
}